// LexicalizedPCFG_62912680952457
// MI455X (gfx1250) — compile-verified
//
#include <hip/hip_runtime.h>

// Lexicalized-PCFG inside algorithm on gfx1250.
// Per (b,head,dir,quadrant) the rule matrix EZ[a,(sl,sr)] is constant across
// spans/splits, so the LSE over symbol pairs becomes an exp-space matmul
//   D(16a x 16m) = EZ(16 x K) * [e^x (x) e^y](K x 16)
// executed as chains of V_WMMA_F32_16X16X4_F32 (f32 keeps log-domain precision).
// Rules (50MB) + charts (16MB) are L2-resident (192MB). Heads are independent,
// so grid = B*L*W single-wave blocks for occupancy; betau reduces afterwards.

typedef __attribute__((ext_vector_type(2))) float v2f;
typedef __attribute__((ext_vector_type(8))) float v8f;

#define NEG (-1.0e30f)

// workspace layout (floats)
#define EZ_SZ    12582912   // B*N*2 * 65536 (swizzled exp'd rules, A-operand layout)
#define QM_SZ    12288      // B*N*2*4quad*16a
#define BETA_SZ  3840000    // [B][25][25][64][24]
#define BETAU_SZ 160000     // [B][25][25][64]

__device__ __forceinline__ int bidx(int b,int l,int r,int s,int h){ return (((b*25+l)*25+r)*64+s)*24+h; }
__device__ __forceinline__ int uidx(int b,int l,int r,int s){ return ((b*25+l)*25+r)*64+s; }
__device__ __forceinline__ int ezbase(int b,int h,int d){ return ((b*24+h)*2+d)*65536; }
__device__ __forceinline__ int qmidx(int b,int h,int d,int q){ return (((b*24+h)*2+d)*4+q)*16; }

__device__ __forceinline__ float lae(float x, float y){
    float m = fmaxf(x, y);
    float n = fminf(x, y);
    return m + log1pf(__expf(n - m));
}

// ---------------- precompute: per-quadrant row maxes of rule scores ----------
__global__ void qmax_kernel(const float* __restrict__ rule, float* __restrict__ qmax){
    int i = blockIdx.x*256 + threadIdx.x;
    if (i >= 12288) return;
    int a = i & 15, q = (i>>4)&3, d = (i>>6)&1, h = (i>>7)%24, b = i/3072;
    int SL0 = (q&2)?16:0, NL=(q&2)?48:16, SR0=(q&1)?16:0, NR=(q&1)?48:16;
    const float* rp = rule + ((((b*16+a)*24+h)*64+SL0)*64+SR0)*2 + d;
    float mx = NEG;
    for (int sl = 0; sl < NL; ++sl)
        for (int sr = 0; sr < NR; ++sr)
            mx = fmaxf(mx, rp[(sl*64+sr)*2]);
    qmax[i] = mx;
}

// ---------------- precompute: EZ = exp(rule - qmax), swizzled as WMMA A ------
// per (b,h,d): 1024 chunks of K=4; chunk c, lane L, vgpr v <-> A[a=L%16, k=4c+v+2*(L/16)]
// quadrant chunk offsets: Q00:0(64) Q01:64(192) Q10:256(192) Q11:448(576)
__global__ void ezs_kernel(const float* __restrict__ rule, const float* __restrict__ qmax,
                           float* __restrict__ ez){
    int i = blockIdx.x*256 + threadIdx.x;   // exactly 12,582,912 threads
    int e   = i & 63;
    int ca  = (i >> 6) & 1023;
    int bhd = i >> 16;
    int d = bhd & 1, h = (bhd >> 1) % 24, b = bhd / 48;
    int L = e >> 1, v = e & 1;
    int q, c;
    if (ca < 64)       { q = 0; c = ca; }
    else if (ca < 256) { q = 1; c = ca - 64; }
    else if (ca < 448) { q = 2; c = ca - 256; }
    else               { q = 3; c = ca - 448; }
    int SL0 = (q&2)?16:0, SR0 = (q&1)?16:0, NR = (q&1)?48:16;
    int a = L & 15, half = L >> 4;
    int k = 4*c + v + 2*half;
    int sl = SL0 + k / NR, sr = SR0 + k % NR;
    float z  = rule[((((b*16+a)*24+h)*64+sl)*64+sr)*2 + d];
    float qm = qmax[qmidx(b,h,d,q) + a];
    ez[i] = __expf(z - qm);
}

// ---------------- chart init -------------------------------------------------
__global__ void fill_kernel(float* __restrict__ p, int n){
    int i = blockIdx.x*256 + threadIdx.x;
    if (i < n) p[i] = NEG;
}
__global__ void diag_kernel(const float* __restrict__ unary,
                            float* __restrict__ beta, float* __restrict__ betau){
    int i = blockIdx.x*256 + threadIdx.x;
    if (i >= 4*24*48) return;
    int b = i / (24*48), rem = i % (24*48), k = rem/48, s = 16 + rem%48;
    beta [bidx(b,k,k+1,s,k)] = 0.0f;
    betau[uidx(b,k,k+1,s)]   = unary[(b*24+k)*64 + s];
}

// ---------------- one WMMA batch: <=16 split columns, one quadrant -----------
template<int Q>
__device__ __forceinline__ void run_batch(
    int d, int b, int l, int r, int h, int mstart, int K,
    const float* __restrict__ beta, const float* __restrict__ betau,
    const float* __restrict__ ez,   const float* __restrict__ qmax,
    float* evw /*[16*96]*/, float* shw /*[16]*/, float* qmw /*[16]*/,
    int lane, float* accP /*[8]*/)
{
    constexpr int SL0  = (Q & 2) ? 16 : 0;
    constexpr int NL   = (Q & 2) ? 48 : 16;
    constexpr int SR0  = (Q & 1) ? 16 : 0;
    constexpr int NR   = (Q & 1) ? 48 : 16;
    constexpr int QOFF = (Q == 0) ? 0 : (Q == 1) ? 64 : (Q == 2) ? 256 : 448;
    constexpr int NCH  = (NL * NR) / 4;
    const int half = lane >> 4;
    const int col  = lane & 15;

    // build column vectors e^{x-shL}, e^{y-shR} in LDS
    for (int j = 0; j < K; ++j) {
        int m = mstart + j;
        // head-side / summed-side selection (invalid sides are NEG -> exp = 0)
        float v0 = NEG, v1 = NEG;
        if (lane < NL)
            v0 = (d == 0) ? beta[bidx(b,l,m,SL0+lane,h)] : betau[uidx(b,l,m,SL0+lane)];
        if (NL > 32 && lane < NL-32)
            v1 = (d == 0) ? beta[bidx(b,l,m,SL0+lane+32,h)] : betau[uidx(b,l,m,SL0+lane+32)];
        float mx = fmaxf(v0, v1);
        #pragma unroll
        for (int s = 16; s >= 1; s >>= 1) mx = fmaxf(mx, __shfl_xor(mx, s, 32));
        float shL = (mx > -1e20f) ? mx : 0.0f;
        if (lane < NL)               evw[j*96 + lane]      = __expf(v0 - shL);
        if (NL > 32 && lane < NL-32) evw[j*96 + lane + 32] = __expf(v1 - shL);

        v0 = NEG; v1 = NEG;
        if (lane < NR)
            v0 = (d == 0) ? betau[uidx(b,m,r,SR0+lane)] : beta[bidx(b,m,r,SR0+lane,h)];
        if (NR > 32 && lane < NR-32)
            v1 = (d == 0) ? betau[uidx(b,m,r,SR0+lane+32)] : beta[bidx(b,m,r,SR0+lane+32,h)];
        mx = fmaxf(v0, v1);
        #pragma unroll
        for (int s = 16; s >= 1; s >>= 1) mx = fmaxf(mx, __shfl_xor(mx, s, 32));
        float shR = (mx > -1e20f) ? mx : 0.0f;
        if (lane < NR)               evw[j*96 + NL + lane]      = __expf(v0 - shR);
        if (NR > 32 && lane < NR-32) evw[j*96 + NL + lane + 32] = __expf(v1 - shR);

        if (lane == 0) shw[j] = shL + shR;
    }
    if (lane < 16) qmw[lane] = qmax[qmidx(b,h,d,Q) + lane];
    asm volatile("s_wait_dscnt 0" ::: "memory");   // wave-local LDS visibility

    // D(16a x 16m) = EZ(16 x NCH*4) * B; two independent WMMA chains for ILP
    v8f acc0 = {}, acc1 = {};
    const float* ezb = ez + ezbase(b,h,d) + QOFF*64;
    for (int c = 0; c < NCH; c += 2) {
        v2f A0 = *(const v2f*)(ezb + (c    )*64 + lane*2);   // coalesced b64
        v2f A1 = *(const v2f*)(ezb + (c + 1)*64 + lane*2);
        int p0 = 4*c + 2*half;
        v2f B0, B1;
        B0[0] = evw[col*96 + (p0    )/NR] * evw[col*96 + NL + (p0    )%NR];
        B0[1] = evw[col*96 + (p0 + 1)/NR] * evw[col*96 + NL + (p0 + 1)%NR];
        B1[0] = evw[col*96 + (p0 + 4)/NR] * evw[col*96 + NL + (p0 + 4)%NR];
        B1[1] = evw[col*96 + (p0 + 5)/NR] * evw[col*96 + NL + (p0 + 5)%NR];
        acc0 = __builtin_amdgcn_wmma_f32_16x16x4_f32(false, A0, false, B0, (short)0, acc0, false, false);
        acc1 = __builtin_amdgcn_wmma_f32_16x16x4_f32(false, A1, false, B1, (short)0, acc1, false, false);
    }
    v8f acc = acc0 + acc1;   // partial sums of the same matmul

    // log + restore shifts, fold into per-lane running LSE (lane owns column m)
    if (col < K) {
        float sh = shw[col];
        #pragma unroll
        for (int vv = 0; vv < 8; ++vv) {
            float dd = acc[vv];
            float contrib = (dd > 0.0f) ? (__logf(dd) + sh + qmw[vv + 8*half]) : NEG;
            accP[vv] = lae(accP[vv], contrib);
        }
    }
}

// ---------------- per-(width, head) DP kernel: one wave per block ------------
__global__ __launch_bounds__(32)
void width_kernel(int W,
                  const float* __restrict__ ez, const float* __restrict__ qmax,
                  float* __restrict__ beta, const float* __restrict__ betau)
{
    __shared__ float EV[16*96];
    __shared__ float SH[16];
    __shared__ float QM[16];

    const int L    = 24 - W + 1;
    const int b    = blockIdx.x / (L * W);
    const int rem  = blockIdx.x % (L * W);
    const int l    = rem / W;
    const int w    = rem % W;
    const int r    = l + W;
    const int h    = l + w;
    const int lane = threadIdx.x;

    float accP[8];
    #pragma unroll
    for (int i = 0; i < 8; ++i) accP[i] = NEG;

    for (int d = 0; d < 2; ++d) {
        if (W == 2) {                       // both children terminal
            run_batch<3>(d,b,l,r,h, l+1, 1, beta,betau,ez,qmax, EV,SH,QM, lane,accP);
        } else {
            run_batch<2>(d,b,l,r,h, l+1, 1, beta,betau,ez,qmax, EV,SH,QM, lane,accP); // left terminal
            run_batch<1>(d,b,l,r,h, r-1, 1, beta,betau,ez,qmax, EV,SH,QM, lane,accP); // right terminal
            for (int base = 0; base < W-3; base += 16) {                               // middle splits
                int K = W-3-base; if (K > 16) K = 16;
                run_batch<0>(d,b,l,r,h, l+2+base, K, beta,betau,ez,qmax, EV,SH,QM, lane,accP);
            }
        }
    }
    // butterfly LSE across the 16 column-lanes of each half
    #pragma unroll
    for (int s = 8; s >= 1; s >>= 1)
        #pragma unroll
        for (int vv = 0; vv < 8; ++vv)
            accP[vv] = lae(accP[vv], __shfl_xor(accP[vv], s, 32));
    if ((lane & 15) == 0) {                 // beta[l,r,a,h] = tmp[a, this head]
        int abase = (lane >> 4) * 8;
        #pragma unroll
        for (int vv = 0; vv < 8; ++vv) beta[bidx(b, l, r, abase + vv, h)] = accP[vv];
    }
}

// ---------------- per-width betau reduction over heads -----------------------
__global__ void betau_kernel(int W, const float* __restrict__ unary,
                             const float* __restrict__ beta, float* __restrict__ betau){
    const int L = 24 - W + 1;
    int i = blockIdx.x*256 + threadIdx.x;
    if (i >= 4 * L * 16) return;
    int a = i & 15, rest = i >> 4;
    int l = rest % L, b = rest / L, r = l + W;
    float acc = NEG;
    for (int w = 0; w < W; ++w)
        acc = lae(acc, beta[bidx(b, l, r, a, l + w)] + unary[(b*24 + (l+w))*64 + a]);
    betau[uidx(b, l, r, a)] = acc;
}

// ---------------- root reduction --------------------------------------------
__global__ void final_kernel(const float* __restrict__ betau, const float* __restrict__ root,
                             float* __restrict__ out){
    int b = threadIdx.x;
    if (b >= 4) return;
    float acc = NEG;
    for (int a = 0; a < 16; ++a)
        acc = lae(acc, betau[uidx(b,0,24,a)] + root[b*16 + a]);
    out[b] = acc;
}

extern "C" void kernel_launch(void* const* d_in, const int* in_sizes, int n_in,
                              void* d_out, int out_size, void* d_ws, size_t ws_size,
                              hipStream_t stream)
{
    (void)in_sizes; (void)n_in; (void)out_size; (void)ws_size;
    const float* unary = (const float*)d_in[0];
    const float* rule  = (const float*)d_in[1];
    const float* root  = (const float*)d_in[2];

    float* ws    = (float*)d_ws;
    float* ez    = ws;
    float* qmax  = ez   + EZ_SZ;
    float* beta  = qmax + QM_SZ;
    float* betau = beta + BETA_SZ;

    qmax_kernel<<<48,    256, 0, stream>>>(rule, qmax);
    ezs_kernel <<<49152, 256, 0, stream>>>(rule, qmax, ez);
    fill_kernel<<<15625, 256, 0, stream>>>(beta, BETA_SZ + BETAU_SZ);
    diag_kernel<<<18,    256, 0, stream>>>(unary, beta, betau);
    for (int W = 2; W <= 24; ++W) {
        int L = 24 - W + 1;
        width_kernel<<<4*L*W, 32, 0, stream>>>(W, ez, qmax, beta, betau);
        betau_kernel<<<(4*L*16 + 255)/256, 256, 0, stream>>>(W, unary, beta, betau);
    }
    final_kernel<<<1, 32, 0, stream>>>(betau, root, (float*)d_out);
}